// ResidualBlock_23682449670676
// MI455X (gfx1250) — compile-verified
//
#include <hip/hip_runtime.h>
#include <math.h>

// ---------------------------------------------------------------------------
// ResidualBlock (BN -> leaky -> GINEConv) x2 + residual, D=128, EA=3, H=8
// d_in layout (flattened setup_inputs dict):
//  0 x[N*128] f32, 1 edge_index[2*E] i32, 2 edge_attr[E*3] f32,
//  3 bn1_g, 4 bn1_b, 5 bn2_g, 6 bn2_b  (each [128])
//  conv1: 7 nn1_w[128*128], 8 nn1_b[128], 9 nn2_w[128*128], 10 nn2_b[128],
//         11 et1_w[3*8], 12 et1_b[8], 13 et2_w[8*128], 14 et2_b[128]
//  conv2: 15..22 same order
// ---------------------------------------------------------------------------

typedef float v2f __attribute__((ext_vector_type(2)));
typedef float v8f __attribute__((ext_vector_type(8)));
typedef int   v4i __attribute__((vector_size(16)));   // matches builtin proto

typedef __attribute__((address_space(1))) v4i* gv4i_p;  // global
typedef __attribute__((address_space(3))) v4i* lv4i_p;  // LDS

#define D 128

#if defined(__has_builtin)
#if __has_builtin(__builtin_amdgcn_global_load_async_to_lds_b128)
#define HAVE_ASYNC_LDS 1
#endif
#endif

__device__ __forceinline__ float leaky01(float v) {
    return v >= 0.0f ? v : 0.01f * v;
}

// ---------------- zero the 256-float stats buffer ---------------------------
__global__ void k_zero256(float* __restrict__ p) {
    p[threadIdx.x] = 0.0f;
}

// ---------------- column sum / sum-of-squares over N rows -------------------
__global__ __launch_bounds__(256) void k_colstats(const float* __restrict__ x,
                                                  float* __restrict__ stats,
                                                  int n) {
    int tid = blockIdx.x * blockDim.x + threadIdx.x;
    int c   = tid & (D - 1);
    int rg  = tid >> 7;
    int nrg = (gridDim.x * blockDim.x) >> 7;
    float s = 0.0f, s2 = 0.0f;
    for (int r = rg; r < n; r += nrg) {
        float v = x[(size_t)r * D + c];
        s += v;
        s2 += v * v;
    }
    atomicAdd(&stats[c], s);
    atomicAdd(&stats[D + c], s2);
}

// ---------------- fused BN(train) + leaky; also seeds agg = h ---------------
__global__ __launch_bounds__(256) void k_bn_leaky(const float* __restrict__ x,
                                                  const float* __restrict__ stats,
                                                  const float* __restrict__ gamma,
                                                  const float* __restrict__ beta,
                                                  float* __restrict__ h,
                                                  float* __restrict__ agg,
                                                  int n) {
    int idx = blockIdx.x * blockDim.x + threadIdx.x;
    if (idx >= n * D) return;
    int c       = idx & (D - 1);
    float invN  = 1.0f / (float)n;
    float mean  = stats[c] * invN;
    float var   = stats[D + c] * invN - mean * mean;
    float v     = gamma[c] * (x[idx] - mean) * rsqrtf(var + 1e-5f) + beta[c];
    v           = leaky01(v);
    h[idx]      = v;   // gathered by edge kernel
    agg[idx]    = v;   // GIN self term (1+eps)*h, eps=0
}

// ---------------- edge kernel: gate MLP + gather*gate + scatter-add ---------
// One wave (32 lanes) per edge; lane handles 4 contiguous features.
// Gate tensor [E,128] (256 MB) is never materialized — computed in registers.
__global__ __launch_bounds__(256) void k_edge(const float* __restrict__ h,
                                              const int* __restrict__ ei,
                                              const float* __restrict__ eattr,
                                              const float* __restrict__ et1w,
                                              const float* __restrict__ et1b,
                                              const float* __restrict__ et2w,
                                              const float* __restrict__ et2b,
                                              float* __restrict__ agg,
                                              int nE) {
    int gw = (blockIdx.x * blockDim.x + threadIdx.x) >> 5;  // edge id
    if (gw >= nE) return;
    int lane = threadIdx.x & 31;

    int src = ei[gw];
    int dst = ei[nE + gw];

    float a0 = eattr[gw * 3 + 0];
    float a1 = eattr[gw * 3 + 1];
    float a2 = eattr[gw * 3 + 2];

    float t[8];
#pragma unroll
    for (int j = 0; j < 8; ++j) {
        float v = a0 * et1w[j] + a1 * et1w[8 + j] + a2 * et1w[16 + j] + et1b[j];
        t[j] = leaky01(v);
    }

    int d0 = lane * 4;
    float4 hx = *(const float4*)(h + (size_t)src * D + d0);  // coalesced 512B/wave

    float g[4];
#pragma unroll
    for (int i = 0; i < 4; ++i) {
        int d = d0 + i;
        float s = et2b[d];
#pragma unroll
        for (int j = 0; j < 8; ++j) s += t[j] * et2w[j * D + d];
        g[i] = s > 0.0f ? s : (__expf(s) - 1.0f);  // ELU(alpha=1)
    }

    float* arow = agg + (size_t)dst * D + d0;
    atomicAdd(arow + 0, hx.x * g[0]);
    atomicAdd(arow + 1, hx.y * g[1]);
    atomicAdd(arow + 2, hx.z * g[2]);
    atomicAdd(arow + 3, hx.w * g[3]);
}

// ---------------- node MLP: leaky(A@W1+b1)@W2+b2 (+resid), f32 WMMA ---------
// Weights async-staged to LDS (pad stride 144 floats: row-pair bank shift
// 2*144 mod 64 = 32 -> the two 16-lane halves of each ds_load hit disjoint
// banks). Interm strip stride 132 -> GEMM2 A-loads cover all 64 banks once.
#define MLP_WAVES 4
#define WSTRIDE 144
#define ISTRIDE 132

__device__ __forceinline__ void stage_weights(const float* __restrict__ w,
                                              float* __restrict__ sW) {
    // 128 rows x 32 float4 chunks, padded rows in LDS
    for (int u = threadIdx.x; u < 128 * 32; u += blockDim.x) {
        int r = u >> 5;
        int c = (u & 31) * 4;
#if defined(HAVE_ASYNC_LDS)
        __builtin_amdgcn_global_load_async_to_lds_b128(
            (gv4i_p)(w + (size_t)r * D + c),
            (lv4i_p)(sW + r * WSTRIDE + c),
            0, 0);
#else
        *(float4*)(sW + r * WSTRIDE + c) = *(const float4*)(w + (size_t)r * D + c);
#endif
    }
#if defined(HAVE_ASYNC_LDS)
    asm volatile("s_wait_asynccnt 0" ::: "memory");
#endif
}

__global__ __launch_bounds__(128) void k_gin_mlp(const float* __restrict__ A,
                                                 const float* __restrict__ w1,
                                                 const float* __restrict__ b1,
                                                 const float* __restrict__ w2,
                                                 const float* __restrict__ b2,
                                                 const float* __restrict__ resid,
                                                 float* __restrict__ out,
                                                 int nrows) {
    __shared__ float sW[128 * WSTRIDE];                // 72 KB (w1, then w2)
    __shared__ float sInterm[MLP_WAVES][16 * ISTRIDE]; // 33 KB

    int lane  = threadIdx.x & 31;
    int wv    = threadIdx.x >> 5;
    int strip = blockIdx.x * MLP_WAVES + wv;
    bool active = (strip * 16 < nrows);   // wave-uniform
    int row0 = strip * 16;

    int am = lane & 15;            // A row / B,C col within tile
    int ak = (lane >> 4) << 1;     // k sub-offset: 0 or 2
    int ch = (lane >> 4) << 3;     // C row offset: 0 or 8

    // ---- phase 1: stage w1, GEMM1 = A(strip) @ w1 ----
    stage_weights(w1, sW);
    __syncthreads();

    float* S = &sInterm[wv][0];
    if (active) {
        const float* Arow = A + (size_t)(row0 + am) * D;
        v8f acc[8] = {};
        for (int k = 0; k < D; k += 4) {
            v2f a;
            a.x = Arow[k + ak];
            a.y = Arow[k + ak + 1];
#pragma unroll
            for (int n = 0; n < 8; ++n) {
                v2f bf;
                bf.x = sW[(k + ak) * WSTRIDE + n * 16 + am];
                bf.y = sW[(k + ak + 1) * WSTRIDE + n * 16 + am];
                acc[n] = __builtin_amdgcn_wmma_f32_16x16x4_f32(
                    false, a, false, bf, (short)0, acc[n], false, false);
            }
        }
        // bias + leaky -> per-wave LDS strip (cross-lane re-layout)
#pragma unroll
        for (int n = 0; n < 8; ++n) {
            float bias = b1[n * 16 + am];
#pragma unroll
            for (int v = 0; v < 8; ++v) {
                S[(v + ch) * ISTRIDE + n * 16 + am] = leaky01(acc[n][v] + bias);
            }
        }
    }
    __syncthreads();  // all reads of sW(w1) + interm writes complete

    // ---- phase 2: stage w2, GEMM2 = leaky1 @ w2 ----
    stage_weights(w2, sW);
    __syncthreads();

    if (active) {
        v8f acc2[8] = {};
        for (int k = 0; k < D; k += 4) {
            v2f a;
            a.x = S[am * ISTRIDE + k + ak];
            a.y = S[am * ISTRIDE + k + ak + 1];
#pragma unroll
            for (int n = 0; n < 8; ++n) {
                v2f bf;
                bf.x = sW[(k + ak) * WSTRIDE + n * 16 + am];
                bf.y = sW[(k + ak + 1) * WSTRIDE + n * 16 + am];
                acc2[n] = __builtin_amdgcn_wmma_f32_16x16x4_f32(
                    false, a, false, bf, (short)0, acc2[n], false, false);
            }
        }
        // bias (+ residual) -> out
#pragma unroll
        for (int n = 0; n < 8; ++n) {
            int col = n * 16 + am;
            float bias = b2[col];
#pragma unroll
            for (int v = 0; v < 8; ++v) {
                size_t idx = (size_t)(row0 + v + ch) * D + col;
                float val = acc2[n][v] + bias;
                if (resid) val += resid[idx];
                out[idx] = val;
            }
        }
    }
}

// ---------------------------------------------------------------------------
extern "C" void kernel_launch(void* const* d_in, const int* in_sizes, int n_in,
                              void* d_out, int out_size, void* d_ws, size_t ws_size,
                              hipStream_t stream) {
    const float* x     = (const float*)d_in[0];
    const int*   ei    = (const int*)d_in[1];
    const float* eattr = (const float*)d_in[2];
    const float* bn1g  = (const float*)d_in[3];
    const float* bn1b  = (const float*)d_in[4];
    const float* bn2g  = (const float*)d_in[5];
    const float* bn2b  = (const float*)d_in[6];
    // conv1
    const float* c1_nn1w = (const float*)d_in[7];
    const float* c1_nn1b = (const float*)d_in[8];
    const float* c1_nn2w = (const float*)d_in[9];
    const float* c1_nn2b = (const float*)d_in[10];
    const float* c1_et1w = (const float*)d_in[11];
    const float* c1_et1b = (const float*)d_in[12];
    const float* c1_et2w = (const float*)d_in[13];
    const float* c1_et2b = (const float*)d_in[14];
    // conv2
    const float* c2_nn1w = (const float*)d_in[15];
    const float* c2_nn1b = (const float*)d_in[16];
    const float* c2_nn2w = (const float*)d_in[17];
    const float* c2_nn2b = (const float*)d_in[18];
    const float* c2_et1w = (const float*)d_in[19];
    const float* c2_et1b = (const float*)d_in[20];
    const float* c2_et2w = (const float*)d_in[21];
    const float* c2_et2b = (const float*)d_in[22];

    const int N_ = in_sizes[0] / D;    // 50000
    const int E_ = in_sizes[1] / 2;    // 500000
    const size_t ND = (size_t)N_ * D;

    float* h     = (float*)d_ws;       // [N,128]
    float* agg   = h + ND;             // [N,128]
    float* y     = agg + ND;           // [N,128] conv1 output
    float* stats = y + ND;             // [256]

    const int bnBlocks   = (N_ * D + 255) / 256;
    const int edgeBlocks = (E_ + 7) / 8;                 // 8 waves/block, 1 edge/wave
    const int strips     = (N_ + 15) / 16;
    const int mlpBlocks  = (strips + MLP_WAVES - 1) / MLP_WAVES;

    // ---- layer 1: BN1 + leaky -> GINEConv1 ----
    k_zero256<<<1, 256, 0, stream>>>(stats);
    k_colstats<<<256, 256, 0, stream>>>(x, stats, N_);
    k_bn_leaky<<<bnBlocks, 256, 0, stream>>>(x, stats, bn1g, bn1b, h, agg, N_);
    k_edge<<<edgeBlocks, 256, 0, stream>>>(h, ei, eattr, c1_et1w, c1_et1b,
                                           c1_et2w, c1_et2b, agg, E_);
    k_gin_mlp<<<mlpBlocks, 128, 0, stream>>>(agg, c1_nn1w, c1_nn1b, c1_nn2w,
                                             c1_nn2b, nullptr, y, N_);

    // ---- layer 2: BN2 + leaky -> GINEConv2 -> +x residual ----
    k_zero256<<<1, 256, 0, stream>>>(stats);
    k_colstats<<<256, 256, 0, stream>>>(y, stats, N_);
    k_bn_leaky<<<bnBlocks, 256, 0, stream>>>(y, stats, bn2g, bn2b, h, agg, N_);
    k_edge<<<edgeBlocks, 256, 0, stream>>>(h, ei, eattr, c2_et1w, c2_et1b,
                                           c2_et2w, c2_et2b, agg, E_);
    k_gin_mlp<<<mlpBlocks, 128, 0, stream>>>(agg, c2_nn1w, c2_nn1b, c2_nn2w,
                                             c2_nn2b, x, (float*)d_out, N_);
}